// sRPE_30537217474911
// MI455X (gfx1250) — compile-verified
//
#include <hip/hip_runtime.h>
#include <cmath>

// sRPE forward for MI455X (gfx1250, wave32, WMMA).
// Key structure: pairwise distance d = t-s takes only 2T-1 = 767 values, so the
// per-(t,s) MLP collapses to a 767-row table Rd[u, 256]. Then
//   out[b,h,t,s] = sum_f x[b,h,t,f] * Rd[t-s+383, h*32+f]
// is a batched GEMM (M=384, N=767, K=32) whose result is scattered along
// anti-diagonals. Both GEMMs use V_WMMA_F32_16X16X4_F32 (exact f32 math).

typedef float v2f __attribute__((ext_vector_type(2)));
typedef float v8f __attribute__((ext_vector_type(8)));

#define TK   384      // tokens
#define HH   8        // heads
#define FDM  32       // head dim
#define NE   256      // n_embd
#define NU   768      // padded relative-offset count (valid 0..766)

// ---------------- Stage 1a: hid[u,c] = gelu(feats(u) @ W1.T + b1) -----------
__global__ void srpe_hid_kernel(const float* __restrict__ pd,   // [B,T,T], batch 0 used
                                const float* __restrict__ W1,   // [256,3]
                                const float* __restrict__ b1,   // [256]
                                float* __restrict__ hid)        // [NU,256]
{
    int idx = blockIdx.x * blockDim.x + threadIdx.x;   // u*256 + c
    if (idx >= NU * NE) return;
    int u = idx >> 8;
    int c = idx & 255;
    int ue = (u < 766) ? u : 766;                      // row 767 is padding
    // read actual d = t-s from the input: (t,s) = (ue-383, 0) or (0, 383-ue)
    float d = (ue >= 383) ? pd[(ue - 383) * TK] : pd[383 - ue];
    float f0 = log1pf(fmaxf(d, 0.0f));
    float f1 = log1pf(fmaxf(-d, 0.0f));
    float f2 = (d == 0.0f) ? 1.0f : 0.0f;
    float e  = f0 * W1[c * 3 + 0] + f1 * W1[c * 3 + 1] + f2 * W1[c * 3 + 2] + b1[c];
    // NewGELU (tanh approximation)
    float e3 = e * e * e;
    float th = tanhf(0.79788456080286535588f * (e + 0.044715f * e3));
    hid[idx] = 0.5f * e * (1.0f + th);
}

// ---------------- Stage 1b: Rd = hid @ W2.T + b2  (WMMA f32, K=256) ---------
// One wave per 16x16 tile of Rd [NU x NE]. grid.x = (NU/16)*(NE/16) = 768.
__global__ void srpe_rd_wmma(const float* __restrict__ hid,   // [NU,256]
                             const float* __restrict__ W2,    // [256,256] row-major
                             const float* __restrict__ b2,    // [256]
                             float* __restrict__ Rd)          // [NU,256]
{
    int tile = blockIdx.x;
    int ci   = tile & 15;          // 16 column tiles
    int ui   = tile >> 4;          // 48 row tiles
    int u0   = ui * 16;
    int c0   = ci * 16;
    int lane = threadIdx.x;        // wave32
    int m    = lane & 15;          // row (A) / col (B,C) within tile
    int half = lane >> 4;          // K sub-group for A/B fragments

    v8f acc = {};
    for (int kb = 0; kb < NE; kb += 4) {
        int ka = kb + half * 2;
        v2f a, b;
        // A (16x4): lane m = M, half selects K pair {0,1} vs {2,3}
        a.x = hid[(u0 + m) * NE + ka];
        a.y = hid[(u0 + m) * NE + ka + 1];
        // B (4x16): B[k,n] = W2[(c0+n)*256 + k]  (hid @ W2^T)
        b.x = W2[(c0 + m) * NE + ka];
        b.y = W2[(c0 + m) * NE + ka + 1];
        acc = __builtin_amdgcn_wmma_f32_16x16x4_f32(
                  false, a, false, b, (short)0, acc, false, false);
    }
    float bias = b2[c0 + m];
    #pragma unroll
    for (int r = 0; r < 8; ++r) {
        int row = r + 8 * half;                 // C layout: VGPR r -> M=r / r+8
        Rd[(u0 + row) * NE + (c0 + m)] = acc[r] + bias;
    }
}

// ------- Stage 2: out[b,h,t,s] = sum_f x[b,h,t,f] * Rd[t-s+383, h*32+f] -----
// Per (b,h): GEMM M=T, N=767 (u), K=32, fused anti-diagonal scatter to out.
// Only the 25 u-tiles per t-tile that can hit s in [0,T) are launched:
//   grid.x = 24 t-tiles * 25 band offsets = 600, grid.y = B*H = 16.
__global__ void srpe_qk_wmma(const float* __restrict__ x,    // [B,H,T,32]
                             const float* __restrict__ Rd,   // [NU,256]
                             float* __restrict__ out)        // [B,H,T,T]
{
    int bh = blockIdx.y;           // b*H + h
    int h  = bh & (HH - 1);
    int id = blockIdx.x;
    int ti = id / 25;              // t-tile index 0..23
    int jo = id % 25;              // band offset
    int t0 = ti * 16;
    int u0 = (ti + jo) * 16;       // u0 <= 752, u max = 767 (padded Rd row)

    int lane = threadIdx.x;
    int m    = lane & 15;
    int half = lane >> 4;

    const float* X = x  + (size_t)bh * TK * FDM;   // [T,32]
    const float* R = Rd + h * FDM;                 // column slice h*32.. of [NU,256]

    v8f acc = {};
    #pragma unroll
    for (int kb = 0; kb < FDM; kb += 4) {
        int ka = kb + half * 2;
        v2f a, b;
        a.x = X[(t0 + m) * FDM + ka];
        a.y = X[(t0 + m) * FDM + ka + 1];
        // B[k,n] = Rd[u0+n, h*32+k]
        b.x = R[(u0 + m) * NE + ka];
        b.y = R[(u0 + m) * NE + ka + 1];
        acc = __builtin_amdgcn_wmma_f32_16x16x4_f32(
                  false, a, false, b, (short)0, acc, false, false);
    }

    float* O = out + (size_t)bh * TK * TK;
    int u = u0 + m;
    #pragma unroll
    for (int r = 0; r < 8; ++r) {
        int t = t0 + r + 8 * half;
        int s = t + 383 - u;                    // u = t-s+383
        if (s >= 0 && s < TK) O[t * TK + s] = acc[r];
    }
}

// ---------------------------------------------------------------------------
extern "C" void kernel_launch(void* const* d_in, const int* in_sizes, int n_in,
                              void* d_out, int out_size, void* d_ws, size_t ws_size,
                              hipStream_t stream)
{
    const float* x  = (const float*)d_in[0];   // [2,8,384,32]
    const float* pd = (const float*)d_in[1];   // [2,384,384]
    const float* W1 = (const float*)d_in[2];   // [256,3]
    const float* b1 = (const float*)d_in[3];   // [256]
    const float* W2 = (const float*)d_in[4];   // [256,256]
    const float* b2 = (const float*)d_in[5];   // [256]
    float* out = (float*)d_out;                // [2,8,384,384]

    // workspace: hid [768*256] f32, Rd [768*256] f32  (~1.5 MB total)
    float* hid = (float*)d_ws;
    float* Rd  = hid + NU * NE;

    // Stage 1a: 768*256 elements, 256 threads/block
    srpe_hid_kernel<<<dim3((NU * NE) / 256), dim3(256), 0, stream>>>(pd, W1, b1, hid);

    // Stage 1b: one wave per 16x16 tile of [768 x 256]
    srpe_rd_wmma<<<dim3((NU / 16) * (NE / 16)), dim3(32), 0, stream>>>(hid, W2, b2, Rd);

    // Stage 2: 24 t-tiles * 25 band tiles, one wave each; y = B*H
    srpe_qk_wmma<<<dim3(24 * 25, 16), dim3(32), 0, stream>>>(x, Rd, out);
}